// Contrastive_41300405518992
// MI455X (gfx1250) — compile-verified
//
#include <hip/hip_runtime.h>
#include <math.h>

// ---------------------------------------------------------------------------
// Contrastive edge losses on MI455X (gfx1250, wave32).
//   signal_loss = mean over signal edges of ||xi - xj||^2
//   hinge_loss  = mean over edges of ( pid match ? d2 : relu(0.1 - sqrt(d2+eps))^2 )
// Gather/reduce workload; x (2.4 MB) lives in L2. Deterministic two-stage
// reduction: WMMA f32 16x16x4 wave reduce -> LDS -> per-block partial ->
// single-block tree finalize. No atomics (run-to-run determinism).
// ---------------------------------------------------------------------------

typedef float v2f __attribute__((ext_vector_type(2)));
typedef float v8f __attribute__((ext_vector_type(8)));

#define NBLOCKS 512
#define TPB     256
#define NWAVES  (TPB / 32)

// Exact f32 wave32 sum using V_WMMA_F32_16X16X4_F32 (D = A x ones + 0).
// A layout (16x4 f32): lane L holds A[L%16, K=(L<16?0:2)] in VGPR0 and
// A[L%16, K=(L<16?1:3)] in VGPR1.  With B = all ones:
//   D[m,n] = acc[lane m] + acc[lane m+16]   (independent of n)
// C/D layout: lane l, vgpr v -> D[v + (l<16?0:8), l%16], so summing the 8
// result VGPRs gives rows 0-7 on lanes 0-15 and rows 8-15 on lanes 16-31;
// one xor-16 shuffle completes the reduction (result in every lane).
__device__ __forceinline__ float wave_reduce_sum(float v) {
    v2f a; a[0] = v;    a[1] = 0.0f;
    v2f b; b[0] = 1.0f; b[1] = 1.0f;
    v8f c = {};
    v8f d = __builtin_amdgcn_wmma_f32_16x16x4_f32(
        /*neg_a=*/false, a, /*neg_b=*/false, b,
        /*c_mod=*/(short)0, c, /*reuse_a=*/false, /*reuse_b=*/false);
    float s = ((d[0] + d[1]) + (d[2] + d[3])) + ((d[4] + d[5]) + (d[6] + d[7]));
    s += __shfl_xor(s, 16, 32);
    return s;
}

template <bool HINGE>
__global__ void __launch_bounds__(TPB)
edge_loss_kernel(const float* __restrict__ x,
                 const int*   __restrict__ pid,
                 const int*   __restrict__ src,
                 const int*   __restrict__ dst,
                 int E,
                 float* __restrict__ block_partial) {
    float acc = 0.0f;

    const unsigned stride = gridDim.x * blockDim.x;
    for (unsigned e = blockIdx.x * blockDim.x + threadIdx.x; e < (unsigned)E; e += stride) {
        const int i = src[e];
        const int j = dst[e];

        // 12 floats per row, 48 B stride -> 16 B aligned float4 loads.
        const float4* __restrict__ xi = reinterpret_cast<const float4*>(x + (size_t)i * 12);
        const float4* __restrict__ xj = reinterpret_cast<const float4*>(x + (size_t)j * 12);
        const float4 a0 = xi[0], a1 = xi[1], a2 = xi[2];
        const float4 b0 = xj[0], b1 = xj[1], b2 = xj[2];

        float t0, d2;
        t0 = a0.x - b0.x; d2  = t0 * t0;
        t0 = a0.y - b0.y; d2 += t0 * t0;
        t0 = a0.z - b0.z; d2 += t0 * t0;
        t0 = a0.w - b0.w; d2 += t0 * t0;
        t0 = a1.x - b1.x; d2 += t0 * t0;
        t0 = a1.y - b1.y; d2 += t0 * t0;
        t0 = a1.z - b1.z; d2 += t0 * t0;
        t0 = a1.w - b1.w; d2 += t0 * t0;
        t0 = a2.x - b2.x; d2 += t0 * t0;
        t0 = a2.y - b2.y; d2 += t0 * t0;
        t0 = a2.z - b2.z; d2 += t0 * t0;
        t0 = a2.w - b2.w; d2 += t0 * t0;

        if (HINGE) {
            const bool y = (pid[i] == pid[j]);
            const float dist = sqrtf(d2 + 1e-12f);
            const float th   = fmaxf(0.1f - dist, 0.0f);
            acc += y ? d2 : th * th;   // branchless select; sqrt is cheap vs gathers
        } else {
            acc += d2;
        }
    }

    // Wave-level exact f32 reduction via WMMA (EXEC is all-ones here: full
    // 256-thread blocks, reduction is outside any divergent region).
    const float wsum = wave_reduce_sum(acc);

    __shared__ float lds_wave[NWAVES];
    const unsigned lane = threadIdx.x & 31u;
    const unsigned wid  = threadIdx.x >> 5;
    if (lane == 0) lds_wave[wid] = wsum;
    __syncthreads();

    if (threadIdx.x == 0) {
        float bsum = 0.0f;
        #pragma unroll
        for (int w = 0; w < NWAVES; ++w) bsum += lds_wave[w];  // fixed order
        block_partial[blockIdx.x] = bsum;
    }
}

__global__ void __launch_bounds__(256)
finalize_kernel(const float* __restrict__ partials, int nb,
                int n_sig, int n_knn, int n_rnd,
                float* __restrict__ out) {
    __shared__ float red[256];
    __shared__ float means[3];
    const int counts[3] = {n_sig, n_knn, n_rnd};

    for (int a = 0; a < 3; ++a) {
        float s = 0.0f;
        for (int i = threadIdx.x; i < nb; i += blockDim.x) s += partials[a * nb + i];
        red[threadIdx.x] = s;
        __syncthreads();
        for (int off = 128; off > 0; off >>= 1) {       // fixed-shape tree: deterministic
            if ((int)threadIdx.x < off) red[threadIdx.x] += red[threadIdx.x + off];
            __syncthreads();
        }
        if (threadIdx.x == 0) means[a] = red[0] / (float)counts[a];
        __syncthreads();
    }

    if (threadIdx.x == 0) {
        out[0] = means[0];
        out[1] = means[1];
        out[2] = means[2];
        out[3] = means[0] + means[1] + means[2];
    }
}

extern "C" void kernel_launch(void* const* d_in, const int* in_sizes, int n_in,
                              void* d_out, int out_size, void* d_ws, size_t ws_size,
                              hipStream_t stream) {
    const float* x   = (const float*)d_in[0];
    const int*   pid = (const int*)  d_in[1];
    const int*   sig = (const int*)  d_in[2];   // (2, S) row-major: [src..., dst...]
    const int*   knn = (const int*)  d_in[3];
    const int*   rnd = (const int*)  d_in[4];

    const int S  = in_sizes[2] / 2;
    const int Kn = in_sizes[3] / 2;
    const int R  = in_sizes[4] / 2;

    float* ws = (float*)d_ws;   // [0,NB): signal  [NB,2NB): knn  [2NB,3NB): random

    edge_loss_kernel<false><<<NBLOCKS, TPB, 0, stream>>>(x, pid, sig, sig + S,  S,  ws);
    edge_loss_kernel<true ><<<NBLOCKS, TPB, 0, stream>>>(x, pid, knn, knn + Kn, Kn, ws + NBLOCKS);
    edge_loss_kernel<true ><<<NBLOCKS, TPB, 0, stream>>>(x, pid, rnd, rnd + R,  R,  ws + 2 * NBLOCKS);

    finalize_kernel<<<1, 256, 0, stream>>>(ws, NBLOCKS, S, Kn, R, (float*)d_out);
}